// MoLE_46531675685331
// MI455X (gfx1250) — compile-verified
//
#include <hip/hip_runtime.h>
#include <hip/hip_bf16.h>

typedef float v2f __attribute__((ext_vector_type(2)));
typedef float v8f __attribute__((ext_vector_type(8)));

#define B_    64
#define E_    16
#define IN_   256
#define OUT_  256
#define NPG_  2048
#define MWG   128          // output rows per workgroup (8 waves x 16)
#define KC    32           // K rows staged in LDS per chunk
#define BSTR2 272          // padded LDS pair-row stride in float2 units (bank-conflict free)

// ---------------- Phase 1: mixed[b] = sum_e coeff[b][e] * kernel[e] ----------------
// kernel: (E, IN, OUT) row-major; mixed: (B, IN, OUT) row-major in workspace.
// Each thread owns one (i, o4) float4 column-group of the kernel bank for all 16
// experts (held in registers), then loops all 64 graphs. Kernel bank is read from
// HBM exactly once (4 MB); all reads & writes are coalesced float4s.
__global__ __launch_bounds__(256) void mole_mix(const float* __restrict__ kern,
                                                const float* __restrict__ coeff,
                                                float* __restrict__ mixed) {
    __shared__ float cs[B_ * E_];
    const int tid = threadIdx.x;
    for (int j = tid; j < B_ * E_; j += 256) cs[j] = coeff[j];
    __syncthreads();

    const int idx = blockIdx.x * 256 + tid;   // 0 .. 16383  (IN*OUT/4)
    const int i   = idx >> 6;                 // 0 .. 255
    const int o4  = (idx & 63) << 2;          // 0 .. 252

    float4 kf[E_];
#pragma unroll
    for (int e = 0; e < E_; ++e)
        kf[e] = *(const float4*)(kern + (size_t)e * IN_ * OUT_ + (size_t)i * OUT_ + o4);

    for (int b = 0; b < B_; ++b) {
        float4 acc = make_float4(0.f, 0.f, 0.f, 0.f);
#pragma unroll
        for (int e = 0; e < E_; ++e) {
            const float c = cs[b * E_ + e];
            acc.x += c * kf[e].x;  acc.y += c * kf[e].y;
            acc.z += c * kf[e].z;  acc.w += c * kf[e].w;
        }
        *(float4*)(mixed + (size_t)b * IN_ * OUT_ + (size_t)i * OUT_ + o4) = acc;
    }
}

// ---------------- Phase 2: out[b] = X[b] @ mixed[b] + bias  (f32 WMMA) ----------------
// B tile staged in LDS row-pair interleaved: LDS pair-row p holds, per column,
// the adjacent float2 (B[2p][col], B[2p+1][col]).  A B-operand fragment for one
// WMMA is then a single aligned ds_load_b64 per lane (VGPR0=row K / K+2 by half,
// VGPR1=row K+1 / K+3), with zero repacking movs and zero bank conflicts.
__global__ __launch_bounds__(256) void mole_gemm(const float* __restrict__ X,
                                                 const float* __restrict__ mixed,
                                                 const float* __restrict__ bias,
                                                 float* __restrict__ out) {
    extern __shared__ float Bs[];             // (KC/2) * BSTR2 float2s

    const int tid = threadIdx.x;
    const int b   = blockIdx.x >> 4;          // 2048/128 = 16 row tiles per graph
    const int mt  = blockIdx.x & 15;
    const float* __restrict__ Bm = mixed + (size_t)b * IN_ * OUT_;

    const int wave = tid >> 5;
    const int lane = tid & 31;
    const int half = lane >> 4;               // 0: lanes 0-15, 1: lanes 16-31
    const int l16  = lane & 15;
    const int rowBase = b * NPG_ + mt * MWG + wave * 16;
    // A fragment rows: lanes 0-15 -> M=lane, lanes 16-31 -> M=lane-16.
    // A fragment K: half 0 holds K={0,1}, half 1 holds K={2,3} of each 4-wide step.
    const float* __restrict__ Xrow = X + (size_t)(rowBase + l16) * IN_ + 2 * half;

    v8f acc[16];
#pragma unroll
    for (int n = 0; n < 16; ++n) {
        v8f z = {};
        acc[n] = z;
    }

    for (int kc = 0; kc < IN_ / KC; ++kc) {
        // stage KC rows of B into interleaved-pair LDS layout
        for (int g = tid; g < (KC / 2) * (OUT_ / 4); g += 256) {   // 16*64 = 1024
            const int p  = g >> 6;                 // pair-row 0..15
            const int c4 = (g & 63) << 2;          // col group
            const float* src = Bm + (size_t)(kc * KC + 2 * p) * OUT_ + c4;
            const float4 lo = *(const float4*)(src);          // row 2p
            const float4 hi = *(const float4*)(src + OUT_);   // row 2p+1
            float* dst = Bs + ((size_t)p * BSTR2 + c4) * 2;
            *(float4*)(dst)     = make_float4(lo.x, hi.x, lo.y, hi.y);
            *(float4*)(dst + 4) = make_float4(lo.z, hi.z, lo.w, hi.w);
        }
        __syncthreads();

        for (int kk = 0; kk < KC / 4; ++kk) {
            const v2f a = *(const v2f*)(Xrow + kc * KC + kk * 4);
            const v2f* __restrict__ Brow =
                (const v2f*)Bs + (size_t)(2 * kk + half) * BSTR2 + l16;
#pragma unroll
            for (int n = 0; n < 16; ++n) {
                const v2f bf = Brow[n * 16];       // single ds_load_b64
                acc[n] = __builtin_amdgcn_wmma_f32_16x16x4_f32(
                    false, a, false, bf, (short)0, acc[n], false, false);
            }
        }
        __syncthreads();
    }

    // Epilogue: D layout — VGPR r: lanes 0-15 -> M=r, lanes 16-31 -> M=r+8.
    const int outRow0 = rowBase + 8 * half;
#pragma unroll
    for (int n = 0; n < 16; ++n) {
        const int col = n * 16 + l16;
        const float bv = bias[col];
#pragma unroll
        for (int r = 0; r < 8; ++r) {
            out[(size_t)(outRow0 + r) * OUT_ + col] = acc[n][r] + bv;
        }
    }
}

extern "C" void kernel_launch(void* const* d_in, const int* in_sizes, int n_in,
                              void* d_out, int out_size, void* d_ws, size_t ws_size,
                              hipStream_t stream) {
    const float* inputs = (const float*)d_in[0];   // (N, IN) f32
    const float* coeffs = (const float*)d_in[1];   // (B, E)  f32
    /* d_in[2] = n_node (uniform, unused) */
    const float* kern   = (const float*)d_in[3];   // (E, IN, OUT) f32
    const float* bias   = (const float*)d_in[4];   // (OUT,) f32
    float*       out    = (float*)d_out;           // (N, OUT) f32
    float*       mixed  = (float*)d_ws;            // (B, IN, OUT) f32 = 16 MB scratch

    mole_mix<<<dim3((IN_ * OUT_ / 4) / 256), dim3(256), 0, stream>>>(kern, coeffs, mixed);

    const size_t ldsBytes = (size_t)(KC / 2) * BSTR2 * 2 * sizeof(float);  // 34,816 B
    mole_gemm<<<dim3(B_ * (NPG_ / MWG)), dim3(256), ldsBytes, stream>>>(inputs, mixed,
                                                                        bias, out);
}